// BornIteration_21569325760993
// MI455X (gfx1250) — compile-verified
//
#include <hip/hip_runtime.h>
#include <math.h>
#include <stdint.h>

typedef float    v2f __attribute__((ext_vector_type(2)));
typedef float    v8f __attribute__((ext_vector_type(8)));
typedef uint32_t v4u __attribute__((ext_vector_type(4)));
typedef int      v4i __attribute__((ext_vector_type(4)));
typedef int      v8i __attribute__((ext_vector_type(8)));

#define B_  2
#define H_  128
#define W_  128
#define C_  32

// ---------------------------------------------------------------------------
// TDM descriptor (D#) builder + issue for a simple 2D tile:
//   tile_dim0 elements contiguous (data_size units), tile_dim1 rows with
//   stride0 between rows.  Groups 2/3 zero (<=2D tensor).  ISA ch.8 layout.
// ---------------------------------------------------------------------------
__device__ __forceinline__ void tdm_tile2d(bool store, uint32_t lds_addr,
                                           const void* gaddr,
                                           uint32_t data_size_log,  // 0:1B 1:2B 2:4B 3:8B
                                           uint32_t tile0, uint32_t tile1,
                                           uint32_t stride0)
{
    const uint64_t ga = (uint64_t)(uintptr_t)gaddr;
    v4u g0;
    g0.x = 1u;                                           // count=1, user descriptor
    g0.y = lds_addr;                                     // LDS byte address
    g0.z = (uint32_t)ga;                                 // global_addr[31:0]
    g0.w = (uint32_t)((ga >> 32) & 0x1FFFFFFu) | (2u << 30);  // addr[56:32] | type=2
    v8i g1;
    g1[0] = (int)(data_size_log << 16);                  // wg_mask=0, data_size
    g1[1] = (int)(tile0 << 16);                          // tensor_dim0[15:0] = tile0
    g1[2] = (int)(tile1 << 16);                          // tensor_dim1[15:0] = tile1
    g1[3] = (int)(tile0 << 16);                          // tile_dim0
    g1[4] = (int)tile1;                                  // tile_dim1 (tile_dim2=0)
    g1[5] = (int)stride0;                                // tensor_dim0_stride[31:0]
    g1[6] = 0;
    g1[7] = 0;
    v4i g2 = {0, 0, 0, 0};
    v4i g3 = {0, 0, 0, 0};
#if defined(__clang_major__) && (__clang_major__ >= 23)
    v8i gz = {0, 0, 0, 0, 0, 0, 0, 0};
    if (store) __builtin_amdgcn_tensor_store_from_lds(g0, g1, g2, g3, gz, 0);
    else       __builtin_amdgcn_tensor_load_to_lds(g0, g1, g2, g3, gz, 0);
#else
    if (store) __builtin_amdgcn_tensor_store_from_lds(g0, g1, g2, g3, 0);
    else       __builtin_amdgcn_tensor_load_to_lds(g0, g1, g2, g3, 0);
#endif
}

// ---------------------------------------------------------------------------
// Stockham radix-2 stage, [line][pos] LDS layout (forward-FFT kernel).
// ---------------------------------------------------------------------------
__device__ __forceinline__ void fft_stage(const float2* src, float2* dst,
                                          int nlines, int s, float sign,
                                          int tid, int nthreads)
{
    const int Ns = 1 << s;
    const int nb = nlines << 6;
    for (int j = tid; j < nb; j += nthreads) {
        const int line = j >> 6;
        const int t    = j & 63;
        const int offs = t & (Ns - 1);
        const int d0   = ((t >> s) << (s + 1)) + offs;
        const float ang = sign * 3.14159265358979323846f * (float)offs / (float)Ns;
        float sn, cs;
        __sincosf(ang, &sn, &cs);
        const float2 a = src[line * 128 + t];
        const float2 b = src[line * 128 + t + 64];
        const float btr = b.x * cs - b.y * sn;
        const float bti = b.x * sn + b.y * cs;
        dst[line * 128 + d0]      = make_float2(a.x + btr, a.y + bti);
        dst[line * 128 + d0 + Ns] = make_float2(a.x - btr, a.y - bti);
    }
}

// ---------------------------------------------------------------------------
// Stockham radix-2 stage, transposed [pos][line] layout (16 lines), matching
// the TDM tile fill order; line index fastest -> conflict-free LDS access.
// ---------------------------------------------------------------------------
__device__ __forceinline__ void fft_stage_t(const float2* src, float2* dst,
                                            int s, float sign,
                                            int tid, int nthreads)
{
    const int Ns = 1 << s;
    for (int j = tid; j < 1024; j += nthreads) {
        const int line = j & 15;
        const int t    = j >> 4;
        const int offs = t & (Ns - 1);
        const int d0   = ((t >> s) << (s + 1)) + offs;
        const float ang = sign * 3.14159265358979323846f * (float)offs / (float)Ns;
        float sn, cs;
        __sincosf(ang, &sn, &cs);
        const float2 a = src[t * 16 + line];
        const float2 b = src[(t + 64) * 16 + line];
        const float btr = b.x * cs - b.y * sn;
        const float bti = b.x * sn + b.y * cs;
        dst[d0 * 16 + line]        = make_float2(a.x + btr, a.y + bti);
        dst[(d0 + Ns) * 16 + line] = make_float2(a.x - btr, a.y - bti);
    }
}

// ---------------------------------------------------------------------------
// K1: s = project(k)  -- Gaussian MLP, f32 WMMA 16x16x4 for the two 32x32 GEMMs
// ---------------------------------------------------------------------------
__global__ __launch_bounds__(256) void k_mlp(const float* __restrict__ kin,
                                             const float* __restrict__ W1,
                                             const float* __restrict__ b1,
                                             const float* __restrict__ W2,
                                             const float* __restrict__ b2,
                                             const float* __restrict__ W3,
                                             const float* __restrict__ b3,
                                             float* __restrict__ sout)
{
    __shared__ float sW1[32], sb1[32], sb2[32], sb3[32];
    __shared__ float sW2[1024], sW3[1024];
    __shared__ float hbuf[8][16][33];

    const int tid = threadIdx.x;
    if (tid < 32) { sW1[tid] = W1[tid]; sb1[tid] = b1[tid];
                    sb2[tid] = b2[tid]; sb3[tid] = b3[tid]; }
    for (int t = tid; t < 1024; t += 256) { sW2[t] = W2[t]; sW3[t] = W3[t]; }
    __syncthreads();

    const int wv   = tid >> 5;
    const int lane = tid & 31;
    const int pl   = lane & 15;
    const int kh   = lane >> 4;
    const int pbase = (blockIdx.x * 8 + wv) * 16;

    {
        const float kv = kin[pbase + pl];
        for (int c = kh * 16; c < kh * 16 + 16; ++c) {
            const float z = fmaf(kv, sW1[c], sb1[c]);
            hbuf[wv][pl][c] = __expf(-z * z);
        }
    }
    __syncthreads();

    v8f acc0 = {0, 0, 0, 0, 0, 0, 0, 0};
    v8f acc1 = {0, 0, 0, 0, 0, 0, 0, 0};
#pragma unroll
    for (int kk = 0; kk < 8; ++kk) {
        const int kb = kk * 4 + kh * 2;
        v2f a;  a.x  = hbuf[wv][pl][kb];        a.y  = hbuf[wv][pl][kb + 1];
        v2f bA; bA.x = sW2[kb * 32 + pl];       bA.y = sW2[(kb + 1) * 32 + pl];
        v2f bB; bB.x = sW2[kb * 32 + 16 + pl];  bB.y = sW2[(kb + 1) * 32 + 16 + pl];
        acc0 = __builtin_amdgcn_wmma_f32_16x16x4_f32(false, a, false, bA,
                                                     (short)0, acc0, false, false);
        acc1 = __builtin_amdgcn_wmma_f32_16x16x4_f32(false, a, false, bB,
                                                     (short)0, acc1, false, false);
    }
    __syncthreads();
#pragma unroll
    for (int r = 0; r < 8; ++r) {
        const int m = kh * 8 + r;
        const float z0 = acc0[r] + sb2[pl];
        const float z1 = acc1[r] + sb2[16 + pl];
        hbuf[wv][m][pl]      = __expf(-z0 * z0);
        hbuf[wv][m][16 + pl] = __expf(-z1 * z1);
    }
    __syncthreads();

    v8f acc2 = {0, 0, 0, 0, 0, 0, 0, 0};
    v8f acc3 = {0, 0, 0, 0, 0, 0, 0, 0};
#pragma unroll
    for (int kk = 0; kk < 8; ++kk) {
        const int kb = kk * 4 + kh * 2;
        v2f a;  a.x  = hbuf[wv][pl][kb];        a.y  = hbuf[wv][pl][kb + 1];
        v2f bA; bA.x = sW3[kb * 32 + pl];       bA.y = sW3[(kb + 1) * 32 + pl];
        v2f bB; bB.x = sW3[kb * 32 + 16 + pl];  bB.y = sW3[(kb + 1) * 32 + 16 + pl];
        acc2 = __builtin_amdgcn_wmma_f32_16x16x4_f32(false, a, false, bA,
                                                     (short)0, acc2, false, false);
        acc3 = __builtin_amdgcn_wmma_f32_16x16x4_f32(false, a, false, bB,
                                                     (short)0, acc3, false, false);
    }
#pragma unroll
    for (int r = 0; r < 8; ++r) {
        const int m = kh * 8 + r;
        sout[(pbase + m) * 32 + pl]      = acc2[r] + sb3[pl];
        sout[(pbase + m) * 32 + 16 + pl] = acc3[r] + sb3[16 + pl];
    }
}

// ---------------------------------------------------------------------------
// K2: forward FFT over axes (batch, H); batch butterfly fused into the load.
// ---------------------------------------------------------------------------
__global__ __launch_bounds__(256) void k_fft_bh(const float* __restrict__ x,
                                                float2* __restrict__ out)
{
    __shared__ float2 bufA[16 * 128];
    __shared__ float2 bufB[16 * 128];
    const int tid = threadIdx.x;
    const int w   = blockIdx.x >> 2;
    const int cg  = (blockIdx.x & 3) * 8;

    for (int e = tid; e < 1024; e += 256) {
        const int h = e >> 3, c = e & 7;
        const float a0 = x[((0 * H_ + h) * W_ + w) * C_ + cg + c];
        const float a1 = x[((1 * H_ + h) * W_ + w) * C_ + cg + c];
        bufA[c * 128 + h]       = make_float2(a0 + a1, 0.0f);
        bufA[(8 + c) * 128 + h] = make_float2(a0 - a1, 0.0f);
    }
    float2* src = bufA; float2* dst = bufB;
    for (int s = 0; s < 7; ++s) {
        __syncthreads();
        fft_stage(src, dst, 16, s, -1.0f, tid, 256);
        float2* t2 = src; src = dst; dst = t2;
    }
    __syncthreads();
    for (int e = tid; e < 2048; e += 256) {
        const int c = e & 7, b = (e >> 3) & 1, i = e >> 4;
        out[((b * H_ + i) * W_ + w) * C_ + cg + c] = src[(b * 8 + c) * 128 + i];
    }
}

// ---------------------------------------------------------------------------
// K3: per-frequency complex channel mix (G computed on the fly).
// ---------------------------------------------------------------------------
__global__ __launch_bounds__(256) void k_filter(const float2* __restrict__ in,
                                                const float* __restrict__ k0_raw,
                                                const float* __restrict__ amp_raw,
                                                float2* __restrict__ outc)
{
    __shared__ float sk0[1024];
    __shared__ float samp[1024];
    const int tid = threadIdx.x;
    for (int t = tid; t < 1024; t += 256) {
        const float a = amp_raw[t];
        const float b = k0_raw[t];
        samp[t] = (a > 20.0f) ? a : log1pf(__expf(a));
        sk0[t]  = (b > 20.0f) ? b : log1pf(__expf(b));
    }
    __syncthreads();

    const int lane  = tid & 31;
    const int gwave = blockIdx.x * 8 + (tid >> 5);
    const float TWO_PI = 6.28318530717958647692f;

    for (int it = 0; it < 16; ++it) {
        const int p = it * 1024 + gwave;
        const int i = p >> 7, j = p & 127;
        const float fi = TWO_PI * (float)((i < 64) ? i : i - 128) * (1.0f / 128.0f);
        const float fj = TWO_PI * (float)((j < 64) ? j : j - 128) * (1.0f / 128.0f);
        const float pij = fi * fi + fj * fj;

        const float2 x0 = in[((0 * H_ + i) * W_ + j) * C_ + lane];
        const float2 x1 = in[((1 * H_ + i) * W_ + j) * C_ + lane];

        float a0r = 0.f, a0i = 0.f, a1r = 0.f, a1i = 0.f;
#pragma unroll 8
        for (int c = 0; c < 32; ++c) {
            const float a   = fmaf(samp[c * 32 + lane], pij, -sk0[c * 32 + lane]);
            const float inv = 1.0f / fmaf(a, a, 1.0f);
            const float gr  = a * inv;
            const float gi  = inv;
            const float xr0 = __shfl(x0.x, c, 32);
            const float xi0 = __shfl(x0.y, c, 32);
            const float xr1 = __shfl(x1.x, c, 32);
            const float xi1 = __shfl(x1.y, c, 32);
            a0r = fmaf(xr0, gr, fmaf(-xi0, gi, a0r));
            a0i = fmaf(xr0, gi, fmaf( xi0, gr, a0i));
            a1r = fmaf(xr1, gr, fmaf(-xi1, gi, a1r));
            a1i = fmaf(xr1, gi, fmaf( xi1, gr, a1i));
        }
        outc[((0 * H_ + i) * W_ + j) * C_ + lane] = make_float2(a0r, a0i);
        outc[((1 * H_ + i) * W_ + j) * C_ + lane] = make_float2(a1r, a1i);
    }
}

// ---------------------------------------------------------------------------
// K4: inverse FFT along W.  Tile in/out via Tensor Data Mover (async DMA).
//   Tile: 16 contiguous float2 (channels) x 128 rows (w), row stride 32 f2.
// ---------------------------------------------------------------------------
__global__ __launch_bounds__(256) void k_ifft_w(const float2* __restrict__ in,
                                                float2* __restrict__ out)
{
    __shared__ float2 bufA[128 * 16];
    __shared__ float2 bufB[128 * 16];
    const int tid = threadIdx.x;
    const int b  = blockIdx.x >> 8;
    const int i  = (blockIdx.x >> 1) & 127;
    const int cg = (blockIdx.x & 1) * 16;

    const float2* gsrc = in  + (size_t)((b * H_ + i) * W_) * C_ + cg;
    float2*       gdst = out + (size_t)((b * H_ + i) * W_) * C_ + cg;

    if (tid < 32) {            // wave 0 drives the TDM (EXEC-independent, 1 DMA)
        tdm_tile2d(false, (uint32_t)(uintptr_t)bufA, gsrc, 3u, 16u, 128u, 32u);
        __builtin_amdgcn_s_wait_tensorcnt(0);
    }
    __syncthreads();

    float2* src = bufA; float2* dst = bufB;
    for (int s = 0; s < 7; ++s) {
        fft_stage_t(src, dst, s, +1.0f, tid, 256);
        __syncthreads();
        float2* t2 = src; src = dst; dst = t2;
    }
    // 7 stages (odd) -> result in bufB; barrier above makes it TDM-visible
    if (tid < 32)
        tdm_tile2d(true, (uint32_t)(uintptr_t)bufB, gdst, 3u, 16u, 128u, 32u);
    // S_ENDPGM implicitly waits TENSORcnt -> store completes before kernel end
}

// ---------------------------------------------------------------------------
// K5: inverse FFT along H + real part + 1/(128*128).  TDM tile in, TDM tile
//   out of the scaled real plane (16 f32 x 128 rows, row stride 4096 f32).
// ---------------------------------------------------------------------------
__global__ __launch_bounds__(256) void k_ifft_h_real(const float2* __restrict__ in,
                                                     float* __restrict__ out)
{
    __shared__ float2 bufA[128 * 16];
    __shared__ float2 bufB[128 * 16];
    __shared__ float  realbuf[128 * 16];
    const int tid = threadIdx.x;
    const int b  = blockIdx.x >> 8;
    const int w  = (blockIdx.x >> 1) & 127;
    const int cg = (blockIdx.x & 1) * 16;

    const float2* gsrc = in  + (size_t)((b * H_) * W_ + w) * C_ + cg;
    float*        gdst = out + (size_t)((b * H_) * W_ + w) * C_ + cg;

    if (tid < 32) {
        tdm_tile2d(false, (uint32_t)(uintptr_t)bufA, gsrc, 3u, 16u, 128u, 4096u);
        __builtin_amdgcn_s_wait_tensorcnt(0);
    }
    __syncthreads();

    float2* src = bufA; float2* dst = bufB;
    for (int s = 0; s < 7; ++s) {
        fft_stage_t(src, dst, s, +1.0f, tid, 256);
        __syncthreads();
        float2* t2 = src; src = dst; dst = t2;
    }
    const float scale = 1.0f / 16384.0f;
    for (int e = tid; e < 2048; e += 256)
        realbuf[e] = bufB[e].x * scale;
    __syncthreads();
    if (tid < 32)
        tdm_tile2d(true, (uint32_t)(uintptr_t)realbuf, gdst, 2u, 16u, 128u, 4096u);
}

// ---------------------------------------------------------------------------
// K6: t = D(g2, u) -- b128 streaming: lane (q,r) loads float4 of row cc+r,
//   cols 4q..4q+3; shfl_xor tree folds the 4 row groups; lanes 0-7 store b128.
// ---------------------------------------------------------------------------
__global__ __launch_bounds__(256) void k_dmix(const float* __restrict__ x,
                                              const float* __restrict__ g,
                                              float* __restrict__ out)
{
    const int lane  = threadIdx.x & 31;
    const int q     = lane & 7;
    const int rsel  = lane >> 3;
    const int gwave = blockIdx.x * 8 + (threadIdx.x >> 5);
    for (int it = 0; it < 16; ++it) {
        const int p = it * 2048 + gwave;
        const float4* gp = (const float4*)(g + (size_t)p * 1024);
        const float xv = x[p * 32 + lane];
        if (it + 1 < 16)
            __builtin_prefetch(g + (size_t)p * 1024 + (size_t)2048 * 1024 + lane * 32, 0, 1);
        float4 acc = make_float4(0.f, 0.f, 0.f, 0.f);
#pragma unroll
        for (int cc = 0; cc < 32; cc += 4) {
            const float4 gv = gp[cc * 8 + rsel * 8 + q];     // 512B/wave/step
            const float xb = __shfl(xv, cc + rsel, 32);
            acc.x = fmaf(xb, gv.x, acc.x);
            acc.y = fmaf(xb, gv.y, acc.y);
            acc.z = fmaf(xb, gv.z, acc.z);
            acc.w = fmaf(xb, gv.w, acc.w);
        }
        acc.x += __shfl_xor(acc.x, 8, 32);  acc.y += __shfl_xor(acc.y, 8, 32);
        acc.z += __shfl_xor(acc.z, 8, 32);  acc.w += __shfl_xor(acc.w, 8, 32);
        acc.x += __shfl_xor(acc.x, 16, 32); acc.y += __shfl_xor(acc.y, 16, 32);
        acc.z += __shfl_xor(acc.z, 16, 32); acc.w += __shfl_xor(acc.w, 16, 32);
        if (lane < 8)
            ((float4*)(out + p * 32))[q] = acc;
    }
}

// ---------------------------------------------------------------------------
// K7: out = D(g4, sg) + D(g1, Gt) + D(g3, u), fused single pass, b128 loads.
// ---------------------------------------------------------------------------
__global__ __launch_bounds__(256) void k_final(const float* __restrict__ u,
                                               const float* __restrict__ sg,
                                               const float* __restrict__ gt,
                                               const float* __restrict__ g1,
                                               const float* __restrict__ g3,
                                               const float* __restrict__ g4,
                                               float* __restrict__ out)
{
    const int lane  = threadIdx.x & 31;
    const int q     = lane & 7;
    const int rsel  = lane >> 3;
    const int gwave = blockIdx.x * 8 + (threadIdx.x >> 5);
    for (int it = 0; it < 16; ++it) {
        const int p = it * 2048 + gwave;
        const size_t base = (size_t)p * 1024;
        const float4* g1p = (const float4*)(g1 + base);
        const float4* g3p = (const float4*)(g3 + base);
        const float4* g4p = (const float4*)(g4 + base);
        const float xu = u[p * 32 + lane];
        const float xs = sg[p * 32 + lane];
        const float xg = gt[p * 32 + lane];
        if (it + 1 < 16) {
            const size_t nb = base + (size_t)2048 * 1024;
            __builtin_prefetch(g1 + nb + lane * 32, 0, 1);
            __builtin_prefetch(g3 + nb + lane * 32, 0, 1);
            __builtin_prefetch(g4 + nb + lane * 32, 0, 1);
        }
        float4 acc = make_float4(0.f, 0.f, 0.f, 0.f);
#pragma unroll
        for (int cc = 0; cc < 32; cc += 4) {
            const int gi = cc * 8 + rsel * 8 + q;
            const float4 v4 = g4p[gi];
            const float4 v1 = g1p[gi];
            const float4 v3 = g3p[gi];
            const float bs = __shfl(xs, cc + rsel, 32);
            const float bg = __shfl(xg, cc + rsel, 32);
            const float bu = __shfl(xu, cc + rsel, 32);
            acc.x = fmaf(bs, v4.x, fmaf(bg, v1.x, fmaf(bu, v3.x, acc.x)));
            acc.y = fmaf(bs, v4.y, fmaf(bg, v1.y, fmaf(bu, v3.y, acc.y)));
            acc.z = fmaf(bs, v4.z, fmaf(bg, v1.z, fmaf(bu, v3.z, acc.z)));
            acc.w = fmaf(bs, v4.w, fmaf(bg, v1.w, fmaf(bu, v3.w, acc.w)));
        }
        acc.x += __shfl_xor(acc.x, 8, 32);  acc.y += __shfl_xor(acc.y, 8, 32);
        acc.z += __shfl_xor(acc.z, 8, 32);  acc.w += __shfl_xor(acc.w, 8, 32);
        acc.x += __shfl_xor(acc.x, 16, 32); acc.y += __shfl_xor(acc.y, 16, 32);
        acc.z += __shfl_xor(acc.z, 16, 32); acc.w += __shfl_xor(acc.w, 16, 32);
        if (lane < 8)
            ((float4*)(out + p * 32))[q] = acc;
    }
}

// ---------------------------------------------------------------------------
extern "C" void kernel_launch(void* const* d_in, const int* in_sizes, int n_in,
                              void* d_out, int out_size, void* d_ws, size_t ws_size,
                              hipStream_t stream)
{
    (void)in_sizes; (void)n_in; (void)out_size; (void)ws_size;

    const float* u      = (const float*)d_in[0];
    const float* kk     = (const float*)d_in[1];
    const float* g1     = (const float*)d_in[2];
    const float* g2     = (const float*)d_in[3];
    const float* g3     = (const float*)d_in[4];
    const float* g4     = (const float*)d_in[5];
    const float* W1     = (const float*)d_in[6];
    const float* b1     = (const float*)d_in[7];
    const float* W2     = (const float*)d_in[8];
    const float* b2     = (const float*)d_in[9];
    const float* W3     = (const float*)d_in[10];
    const float* b3     = (const float*)d_in[11];
    const float* k0_G   = (const float*)d_in[12];
    const float* amp_G  = (const float*)d_in[13];
    const float* k0_Gs  = (const float*)d_in[14];
    const float* amp_Gs = (const float*)d_in[15];
    float* out = (float*)d_out;

    float*  buf0 = (float*)d_ws;                        // s, later sg
    float*  buf1 = buf0 + (1u << 20);                   // t, later Gt
    float2* c1   = (float2*)(buf1 + (1u << 20));
    float2* c2   = c1 + (1u << 20);

    // s = project(k)
    k_mlp<<<256, 256, 0, stream>>>(kk, W1, b1, W2, b2, W3, b3, buf0);
    // sg = greens(s, k0_Gs, amp_Gs)
    k_fft_bh<<<512, 256, 0, stream>>>(buf0, c1);
    k_filter<<<128, 256, 0, stream>>>(c1, k0_Gs, amp_Gs, c2);
    k_ifft_w<<<512, 256, 0, stream>>>(c2, c1);
    k_ifft_h_real<<<512, 256, 0, stream>>>(c1, buf0);
    // t = D(g2, u)
    k_dmix<<<256, 256, 0, stream>>>(u, g2, buf1);
    // Gt = greens(t, k0_G, amp_G)
    k_fft_bh<<<512, 256, 0, stream>>>(buf1, c1);
    k_filter<<<128, 256, 0, stream>>>(c1, k0_G, amp_G, c2);
    k_ifft_w<<<512, 256, 0, stream>>>(c2, c1);
    k_ifft_h_real<<<512, 256, 0, stream>>>(c1, buf1);
    // out = D(g4, sg) + D(g1, Gt) + D(g3, u)
    k_final<<<256, 256, 0, stream>>>(u, buf0, buf1, g1, g3, g4, out);
}